// MedianBlur_62234076119410
// MI455X (gfx1250) — compile-verified
//
#include <hip/hip_runtime.h>
#include <cstdint>
#include <cstddef>

// ---------------------------------------------------------------------------
// 5x5 exact median blur, f32, zero padding (kornia-compatible).
// CDNA5 (gfx1250) path: async global->LDS tile staging (ASYNCcnt) + wave32
// forgetful-selection median (168 compare-exchanges per pixel).
// ---------------------------------------------------------------------------

#define KSZ 5
#define RAD 2
#define TX  32
#define TY  8
#define HX  (TX + 2 * RAD)   // 36
#define HY  (TY + 2 * RAD)   // 12
#define HALO_N (HX * HY)     // 432 floats
#define NTHREADS (TX * TY)   // 256 threads = 8 wave32

typedef __attribute__((address_space(1))) int GlobalInt;
typedef __attribute__((address_space(3))) int LdsInt;
typedef __attribute__((address_space(3))) float LdsFloat;

// ---- async global -> LDS copy (CDNA5 GLOBAL_LOAD_ASYNC_TO_LDS_B32) --------
__device__ __forceinline__ void async_load_f32(const float* gsrc, float* lds_dst) {
#if defined(__gfx1250__) && __has_builtin(__builtin_amdgcn_global_load_async_to_lds_b32)
  __builtin_amdgcn_global_load_async_to_lds_b32(
      (GlobalInt*)gsrc, (LdsInt*)lds_dst, /*imm offset*/ 0, /*cpol*/ 0);
#else
  // GV mode: vdst = LDS byte address, vaddr = 64-bit global address.
  unsigned lds_addr = (unsigned)(uintptr_t)(LdsFloat*)lds_dst;
  asm volatile("global_load_async_to_lds_b32 %0, %1, off"
               :: "v"(lds_addr), "v"(gsrc)
               : "memory");
#endif
}

__device__ __forceinline__ void wait_async0() {
#if defined(__gfx1250__) && __has_builtin(__builtin_amdgcn_s_wait_asynccnt)
  __builtin_amdgcn_s_wait_asynccnt(0);
#else
  asm volatile("s_wait_asynccnt 0" ::: "memory");
#endif
}

// ---- branchless compare-exchange -------------------------------------------
__device__ __forceinline__ void cas(float& a, float& b) {
  float lo = fminf(a, b);
  float hi = fmaxf(a, b);
  a = lo;
  b = hi;
}

// One forgetful-selection round over v[0..L-1]:
// forward bubble parks max at v[L-1], backward bubble parks min at v[0].
// Cost: 2L-3 compare-exchanges.
template <int L>
__device__ __forceinline__ void minmaxpass(float* v) {
#pragma unroll
  for (int i = 0; i < L - 1; ++i) cas(v[i], v[i + 1]);
#pragma unroll
  for (int i = L - 2; i >= 1; --i) cas(v[i - 1], v[i]);
}

__global__ __launch_bounds__(NTHREADS)
void MedianBlur5x5_kernel(const float* __restrict__ img,
                          float* __restrict__ out,
                          int H, int W) {
  __shared__ float tile[HALO_N];

  const int tx = threadIdx.x;
  const int ty = threadIdx.y;
  const int t  = ty * TX + tx;
  const int x0 = blockIdx.x * TX;
  const int y0 = blockIdx.y * TY;
  const int plane = blockIdx.z;

  const float* src = img + (size_t)plane * H * W;

  // 1) Zero-fill halo tile (provides the zero padding at image borders).
#pragma unroll
  for (int l = t; l < HALO_N; l += NTHREADS) tile[l] = 0.0f;
  __syncthreads();  // DS stores committed before async writes can land

  // 2) Async-gather in-bounds halo elements straight into LDS (ASYNCcnt).
#pragma unroll
  for (int l = t; l < HALO_N; l += NTHREADS) {
    const int lx = l % HX;
    const int ly = l / HX;
    const int gx = x0 + lx - RAD;
    const int gy = y0 + ly - RAD;
    if (gx >= 0 && gx < W && gy >= 0 && gy < H) {
      async_load_f32(src + (size_t)gy * W + gx, &tile[l]);
    }
  }
  wait_async0();    // this wave's async LDS writes are complete
  __syncthreads();  // ...and visible to all waves in the workgroup

  // 3) Read the 5x5 window (conflict-free: lanes hit consecutive DWORDs).
  float w[KSZ * KSZ];
#pragma unroll
  for (int dy = 0; dy < KSZ; ++dy) {
#pragma unroll
    for (int dx = 0; dx < KSZ; ++dx) {
      w[dy * KSZ + dx] = tile[(ty + dy) * HX + (tx + dx)];
    }
  }

  // 4) Forgetful selection: exact median of 25 in 168 compare-exchanges.
  //    Buffer of 14; each round discards the provably-non-median min & max,
  //    then inserts the next unseen value into the freed min slot.
  float v[14];
#pragma unroll
  for (int i = 0; i < 14; ++i) v[i] = w[i];

  minmaxpass<14>(v); v[0] = w[14];
  minmaxpass<13>(v); v[0] = w[15];
  minmaxpass<12>(v); v[0] = w[16];
  minmaxpass<11>(v); v[0] = w[17];
  minmaxpass<10>(v); v[0] = w[18];
  minmaxpass< 9>(v); v[0] = w[19];
  minmaxpass< 8>(v); v[0] = w[20];
  minmaxpass< 7>(v); v[0] = w[21];
  minmaxpass< 6>(v); v[0] = w[22];
  minmaxpass< 5>(v); v[0] = w[23];
  minmaxpass< 4>(v); v[0] = w[24];
  // Live candidates: v[0], v[1], v[2]. Median of 25 == median of these 3.
  const float med =
      fmaxf(fminf(v[0], v[1]), fminf(fmaxf(v[0], v[1]), v[2]));

  const int ox = x0 + tx;
  const int oy = y0 + ty;
  if (ox < W && oy < H) {
    out[(size_t)plane * H * W + (size_t)oy * W + ox] = med;
  }
}

// ---------------------------------------------------------------------------
extern "C" void kernel_launch(void* const* d_in, const int* in_sizes, int n_in,
                              void* d_out, int out_size, void* d_ws, size_t ws_size,
                              hipStream_t stream) {
  (void)n_in; (void)out_size; (void)d_ws; (void)ws_size;

  const float* img = (const float*)d_in[0];
  float* out = (float*)d_out;

  const int H = 512;
  const int W = 512;
  const int planes = in_sizes[0] / (H * W);  // B*C = 24

  dim3 block(TX, TY, 1);
  dim3 grid((W + TX - 1) / TX, (H + TY - 1) / TY, planes);
  hipLaunchKernelGGL(MedianBlur5x5_kernel, grid, block, 0, stream, img, out, H, W);
}